// EnhancedFashionRecommender_75110388073140
// MI455X (gfx1250) — compile-verified
//
#include <hip/hip_runtime.h>
#include <hip/hip_bf16.h>
#include <math.h>

typedef __attribute__((ext_vector_type(2))) float v2f;
typedef __attribute__((ext_vector_type(8))) float v8f;

// Problem constants (match reference)
#define BATCH 8192
#define SEQ 50
#define PROTO 16
#define DIM 128
#define NUSERS 100000

// ---------------------------------------------------------------------------
// Kernel 1: bulk copy user_prototypes -> new_user_prototypes (float4 = B128)
// ---------------------------------------------------------------------------
__global__ void copy_f4_kernel(const float4* __restrict__ src,
                               float4* __restrict__ dst, long n4) {
    long i = (long)blockIdx.x * blockDim.x + threadIdx.x;
    long stride = (long)gridDim.x * blockDim.x;
    for (; i < n4; i += stride) {
        dst[i] = src[i];
    }
}

// ---------------------------------------------------------------------------
// Kernel 2: copy interaction_count -> new_interaction_count
// ---------------------------------------------------------------------------
__global__ void copy_f_kernel(const float* __restrict__ src,
                              float* __restrict__ dst, int n) {
    int i = blockIdx.x * blockDim.x + threadIdx.x;
    if (i < n) dst[i] = src[i];
}

// ---------------------------------------------------------------------------
// Kernel 3: per-user work.  One block (128 threads = 4 waves) per batch row.
//  - masked mean over SEQ, L2 normalize (LDS tree reduce)
//  - curriculum blend (combined), momentum blend (new_rows)
//  - scatter new_rows into the copied prototype table (indices unique: arange)
//  - scatter cnt+1
//  - stash normalized upd into workspace for the WMMA batch reduction
// ---------------------------------------------------------------------------
__global__ __launch_bounds__(DIM) void per_user_kernel(
    const int*   __restrict__ user_idx,
    const float* __restrict__ features,          // [B,S,D]
    const float* __restrict__ success_mask,      // [B,S]
    const float* __restrict__ user_prototypes,   // [NU,P,D]
    const float* __restrict__ shared_prototypes, // [1,P,D]
    const float* __restrict__ interaction_count, // [NU]
    float* __restrict__ out_combined,            // [B,P,D]
    float* __restrict__ out_new_rows,            // [B,P,D]
    float* __restrict__ out_nup,                 // [NU,P,D] (already copied)
    float* __restrict__ out_cnt,                 // [NU] (already copied)
    float* __restrict__ ws_upd)                  // [B,D]
{
    const int b = blockIdx.x;
    const int d = threadIdx.x;

    __shared__ float sm[SEQ];
    __shared__ float red[DIM];

    if (d < SEQ) sm[d] = success_mask[(size_t)b * SEQ + d];
    __syncthreads();

    // denom = max(sum(mask), 1e-6)  (redundant per-thread LDS broadcast read)
    float denom = 0.0f;
#pragma unroll
    for (int s = 0; s < SEQ; ++s) denom += sm[s];
    denom = fmaxf(denom, 1e-6f);

    // masked sum over SEQ, coalesced across d
    const float* f = features + (size_t)b * SEQ * DIM + d;
    float acc = 0.0f;
#pragma unroll 5
    for (int s = 0; s < SEQ; ++s) acc = fmaf(f[(size_t)s * DIM], sm[s], acc);
    float upd = acc / denom;

    // L2 norm across the 128 dims via LDS tree reduce
    red[d] = upd * upd;
    __syncthreads();
    for (int off = DIM / 2; off > 0; off >>= 1) {
        if (d < off) red[d] += red[d + off];
        __syncthreads();
    }
    const float nrm = sqrtf(red[0]);
    upd = upd / fmaxf(nrm, 1e-12f);

    ws_upd[(size_t)b * DIM + d] = upd;

    const int u = user_idx[b];
    const float cnt = interaction_count[u];
    const float w = 1.0f / (1.0f + expf(-(cnt * 0.01f - 3.0f)));     // sigmoid
    const float m = fminf(fmaxf(0.9f + cnt * 0.001f, 0.9f), 0.99f);  // clip

    const float* up_base = user_prototypes + (size_t)u * (PROTO * DIM) + d;
    float* comb = out_combined + (size_t)b * (PROTO * DIM) + d;
    float* nr   = out_new_rows + (size_t)b * (PROTO * DIM) + d;
    float* nup  = out_nup      + (size_t)u * (PROTO * DIM) + d;

#pragma unroll
    for (int p = 0; p < PROTO; ++p) {
        const float up = up_base[p * DIM];
        const float sh = shared_prototypes[p * DIM + d];
        comb[p * DIM] = w * up + (1.0f - w) * sh;
        const float nv = m * up + (1.0f - m) * upd;
        nr[p * DIM]  = nv;
        nup[p * DIM] = nv;  // unique indices -> plain scatter store
    }

    if (d == 0) out_cnt[u] = cnt + 1.0f;
}

// ---------------------------------------------------------------------------
// Kernel 4: batch reduction of upd via V_WMMA_F32_16X16X4_F32.
// mean[d] = (1/B) sum_b upd[b,d]  ==  ones[16x4] x upd_chunk[4x16] accumulated.
// With A = all-ones, D[m,n] = sum_k B[k,n] regardless of K ordering in the
// B-operand register layout.  One full wave32 per block (EXEC all 1s).
// Block g reduces rows [g*256, (g+1)*256) -> partial[g, 0..127].
// ---------------------------------------------------------------------------
__global__ __launch_bounds__(32) void reduce_wmma_kernel(
    const float* __restrict__ upd,      // [B, DIM]
    float* __restrict__ partial)        // [32, DIM]
{
    const int g    = blockIdx.x;    // 0..31
    const int lane = threadIdx.x;   // 0..31
    const bool lo  = lane < 16;
    const int col_lane = lo ? lane : (lane - 16);
    const int base_row = g * 256;

    v2f a;                           // 16x4 all-ones A matrix
    a.x = 1.0f; a.y = 1.0f;

    for (int ct = 0; ct < DIM / 16; ++ct) {
        v8f c = {};
        const int col = ct * 16 + col_lane;
        for (int kk = 0; kk < 64; ++kk) {
            // B is 4x16 f32: lanes 0-15 carry two K rows, lanes 16-31 the
            // other two (exact K assignment irrelevant since A == ones).
            const int r0 = base_row + kk * 4 + (lo ? 0 : 2);
            v2f bm;
            bm.x = upd[(size_t)r0 * DIM + col];
            bm.y = upd[(size_t)(r0 + 1) * DIM + col];
            c = __builtin_amdgcn_wmma_f32_16x16x4_f32(
                    /*neg_a=*/false, a, /*neg_b=*/false, bm,
                    /*c_mod=*/(short)0, c, /*reuse_a=*/false, /*reuse_b=*/false);
        }
        // D row M=0 lives in acc VGPR0, lanes 0..15 (N = lane)
        if (lo) partial[(size_t)g * DIM + ct * 16 + lane] = c[0];
    }
}

// ---------------------------------------------------------------------------
// Kernel 5: finish shared-prototype update.
// ---------------------------------------------------------------------------
__global__ __launch_bounds__(DIM) void finalize_shared_kernel(
    const float* __restrict__ partial,            // [32, DIM]
    const float* __restrict__ shared_prototypes,  // [1,P,D]
    float* __restrict__ out_shared)               // [1,P,D]
{
    const int d = threadIdx.x;
    float s = 0.0f;
#pragma unroll
    for (int g = 0; g < 32; ++g) s += partial[g * DIM + d];
    const float mean = s / (float)BATCH;

    __shared__ float red[DIM];
    red[d] = mean * mean;
    __syncthreads();
    for (int off = DIM / 2; off > 0; off >>= 1) {
        if (d < off) red[d] += red[d + off];
        __syncthreads();
    }
    const float nrm = sqrtf(red[0]);
    const float su = mean / fmaxf(nrm, 1e-12f);

#pragma unroll
    for (int p = 0; p < PROTO; ++p)
        out_shared[p * DIM + d] =
            0.9f * shared_prototypes[p * DIM + d] + 0.1f * su;
}

// ---------------------------------------------------------------------------
extern "C" void kernel_launch(void* const* d_in, const int* in_sizes, int n_in,
                              void* d_out, int out_size, void* d_ws, size_t ws_size,
                              hipStream_t stream) {
    const int*   user_idx          = (const int*)  d_in[0];
    const float* features          = (const float*)d_in[1];
    const float* success_mask      = (const float*)d_in[2];
    const float* user_prototypes   = (const float*)d_in[3];
    const float* shared_prototypes = (const float*)d_in[4];
    const float* interaction_count = (const float*)d_in[5];

    float* out = (float*)d_out;
    const long o_comb = 0;
    const long o_nr   = o_comb + (long)BATCH * PROTO * DIM;   // 16,777,216
    const long o_nup  = o_nr   + (long)BATCH * PROTO * DIM;   // 33,554,432
    const long o_sh   = o_nup  + (long)NUSERS * PROTO * DIM;  // 238,354,432
    const long o_cnt  = o_sh   + (long)PROTO * DIM;           // 238,356,480

    float* ws_upd     = (float*)d_ws;                         // [B, DIM]
    float* ws_partial = ws_upd + (long)BATCH * DIM;           // [32, DIM]

    // 1) bulk copy prototypes (float4 wide)
    {
        const long n4 = (long)NUSERS * PROTO * DIM / 4;       // 51,200,000
        copy_f4_kernel<<<4096, 256, 0, stream>>>(
            (const float4*)user_prototypes, (float4*)(out + o_nup), n4);
    }
    // 2) copy interaction counts
    copy_f_kernel<<<(NUSERS + 255) / 256, 256, 0, stream>>>(
        interaction_count, out + o_cnt, NUSERS);

    // 3) per-user fused kernel (also scatters over the copies above)
    per_user_kernel<<<BATCH, DIM, 0, stream>>>(
        user_idx, features, success_mask, user_prototypes, shared_prototypes,
        interaction_count, out + o_comb, out + o_nr, out + o_nup, out + o_cnt,
        ws_upd);

    // 4) WMMA ones-matvec reduction over the batch
    reduce_wmma_kernel<<<32, 32, 0, stream>>>(ws_upd, ws_partial);

    // 5) finalize shared prototypes
    finalize_shared_kernel<<<1, DIM, 0, stream>>>(
        ws_partial, shared_prototypes, out + o_sh);
}